// MFEST_WIG_13408887898576
// MI455X (gfx1250) — compile-verified
//
#include <hip/hip_runtime.h>
#include <hip/hip_bf16.h>
#include <math.h>

// ---------------------------------------------------------------------------
// Problem constants (match the reference)
// ---------------------------------------------------------------------------
constexpr int kB   = 4;
constexpr int kN   = 256;   // nodes
constexpr int kT   = 64;    // time steps
constexpr int kP   = 6;     // raw feature dim
constexpr int kH   = 128;   // hidden dim
constexpr int kL   = 3;     // conv taps
constexpr int kDOM = 12;    // kept spectral bins
constexpr int kBN  = kB * kN;          // 1024 sequences
constexpr float kPI = 3.14159265358979323846f;

typedef __attribute__((ext_vector_type(2))) float v2f;
typedef __attribute__((ext_vector_type(8))) float v8f;

// fp32 WMMA: D(16x16) += A(16x4) * B(4x16), wave32, fp32 accuracy preserved.
__device__ __forceinline__ v8f wmma_f32(v2f a, v2f b, v8f c) {
  return __builtin_amdgcn_wmma_f32_16x16x4_f32(
      /*neg_a=*/false, a, /*neg_b=*/false, b,
      /*c_mod=*/(short)0, c, /*reuse_a=*/false, /*reuse_b=*/false);
}

// ---------------------------------------------------------------------------
// 1) Spectral low-pass block: per (seq, channel) time series of length T=64.
// ---------------------------------------------------------------------------
__global__ __launch_bounds__(128) void spectral_kernel(
    const float* __restrict__ x, float* __restrict__ out,
    const float* __restrict__ fwr, const float* __restrict__ fwi,
    const float* __restrict__ fbr, const float* __restrict__ fbi, int C) {
  const int seq = blockIdx.x;
  const int c = threadIdx.x;
  if (c >= C) return;

  const float* xp = x + (size_t)seq * kT * C + c;
  float xs[kT];
  float m = 0.f;
#pragma unroll
  for (int t = 0; t < kT; ++t) { xs[t] = xp[(size_t)t * C]; m += xs[t]; }
  m *= (1.f / kT);
  float var = 0.f;
#pragma unroll
  for (int t = 0; t < kT; ++t) { float d = xs[t] - m; var += d * d; }
  var = var / (kT - 1) + 1e-5f;
  const float s  = sqrtf(var);
  const float rs = 1.f / s;

  // DFT bins 1..11 of (x-m)/s.  Bin 0 of the centered signal is exactly 0,
  // and sum(cos)=sum(sin)=0 for f>=1, so run the DFT on raw samples, scale 1/s.
  float Xr[kDOM], Xi[kDOM];
  Xr[0] = 0.f; Xi[0] = 0.f;
  for (int f = 1; f < kDOM; ++f) {
    float ar = 0.f, ai = 0.f;
    const float w = -2.f * kPI * (float)f / (float)kT;
#pragma unroll
    for (int t = 0; t < kT; ++t) {
      float sn, cs;
      __sincosf(w * (float)t, &sn, &cs);
      ar += xs[t] * cs;
      ai += xs[t] * sn;
    }
    Xr[f] = ar * rs;
    Xi[f] = ai * rs;
  }

  // 12x12 complex linear map.
  float our[kDOM], oui[kDOM];
  for (int f = 0; f < kDOM; ++f) {
    float r = fbr[f], i2 = fbi[f];
#pragma unroll
    for (int g = 0; g < kDOM; ++g) {
      const float wr = fwr[f * kDOM + g];
      const float wi = fwi[f * kDOM + g];
      r  += Xr[g] * wr - Xi[g] * wi;
      i2 += Xr[g] * wi + Xi[g] * wr;
    }
    our[f] = r; oui[f] = i2;
  }

  // Inverse rfft with only bins 0..11 populated (irfft drops Im of bin 0).
  float* op = out + (size_t)seq * kT * C + c;
  for (int t = 0; t < kT; ++t) {
    float acc = our[0];
    const float w = 2.f * kPI * (float)t / (float)kT;
#pragma unroll
    for (int f = 1; f < kDOM; ++f) {
      float sn, cs;
      __sincosf(w * (float)f, &sn, &cs);
      acc += 2.f * (our[f] * cs - oui[f] * sn);
    }
    const float low = acc * (1.f / kT);
    op[(size_t)t * C] = low * s + m;
  }
}

// ---------------------------------------------------------------------------
// 2) Generic fp32 WMMA GEMM:
//    C[i,j] = sum_k A[i*lda + k] * B[j*sbj + k*sbk] + bias[j]
//    BVEC: sbk==1 (B fragment contiguous -> 8B vector loads).
//    K loop: step-16 main (4 WMMAs, no predication), step-4 mid, and a
//    branch-free clamped tail for K%4 (uniform scalar branch, cndmask zeros).
// ---------------------------------------------------------------------------
template <bool BVEC>
__device__ __forceinline__ v2f load_bfrag(const float* Bcol, int k0, int sbk) {
  if constexpr (BVEC) {
    return *(const v2f*)(Bcol + k0);
  } else {
    v2f b;
    b.x = Bcol[(size_t)k0 * sbk];
    b.y = Bcol[(size_t)(k0 + 1) * sbk];
    return b;
  }
}

template <bool BVEC>
__global__ __launch_bounds__(256) void wmma_gemm_kernel(
    const float* __restrict__ A, int lda,
    const float* __restrict__ Bm, int sbj, int sbk,
    const float* __restrict__ bias,
    float* __restrict__ Cm, int ldc, int M, int Nc, int K) {
  const int wave = (blockIdx.x * blockDim.x + threadIdx.x) >> 5;
  const int lane = threadIdx.x & 31;
  const int tiles_n = Nc >> 4;
  const int total = (M >> 4) * tiles_n;
  if (wave >= total) return;

  const int m0 = (wave / tiles_n) << 4;
  const int j0 = (wave % tiles_n) << 4;
  const int r  = lane & 15;            // A row / B col within tile
  const int kh = (lane >> 4) << 1;     // K sub-offset: lanes 0-15 {0,1}, 16-31 {2,3}

  const float* Arow = A + (size_t)(m0 + r) * lda;
  const float* Bcol = Bm + (size_t)(j0 + r) * sbj;

  v8f acc = {0.f, 0.f, 0.f, 0.f, 0.f, 0.f, 0.f, 0.f};
  const int K16 = K & ~15;
  const int K4  = K & ~3;
  int k = 0;
  for (; k < K16; k += 16) {
    const v2f a0 = *(const v2f*)(Arow + k + kh);
    const v2f a1 = *(const v2f*)(Arow + k + 4 + kh);
    const v2f a2 = *(const v2f*)(Arow + k + 8 + kh);
    const v2f a3 = *(const v2f*)(Arow + k + 12 + kh);
    const v2f b0 = load_bfrag<BVEC>(Bcol, k + kh, sbk);
    const v2f b1 = load_bfrag<BVEC>(Bcol, k + 4 + kh, sbk);
    const v2f b2 = load_bfrag<BVEC>(Bcol, k + 8 + kh, sbk);
    const v2f b3 = load_bfrag<BVEC>(Bcol, k + 12 + kh, sbk);
    acc = wmma_f32(a0, b0, acc);
    acc = wmma_f32(a1, b1, acc);
    acc = wmma_f32(a2, b2, acc);
    acc = wmma_f32(a3, b3, acc);
  }
  for (; k < K4; k += 4) {
    const v2f a = *(const v2f*)(Arow + k + kh);
    const v2f b = load_bfrag<BVEC>(Bcol, k + kh, sbk);
    acc = wmma_f32(a, b, acc);
  }
  if (k < K) {  // uniform branch; lane predication via index-clamp + cndmask
    const int k0 = k + kh, k1 = k0 + 1;
    const int c0 = (k0 < K) ? k0 : 0;
    const int c1 = (k1 < K) ? k1 : 0;
    v2f a, b;
    a.x = Arow[c0];                 a.y = Arow[c1];
    b.x = Bcol[(size_t)c0 * sbk];   b.y = Bcol[(size_t)c1 * sbk];
    a.x = (k0 < K) ? a.x : 0.f;     a.y = (k1 < K) ? a.y : 0.f;
    b.x = (k0 < K) ? b.x : 0.f;     b.y = (k1 < K) ? b.y : 0.f;
    acc = wmma_f32(a, b, acc);
  }

  const float bb = bias ? bias[j0 + r] : 0.f;
  const int mbase = m0 + ((lane >> 4) << 3);
#pragma unroll
  for (int i = 0; i < 8; ++i) {
    Cm[(size_t)(mbase + i) * ldc + j0 + r] = acc[i] + bb;
  }
}

// ---------------------------------------------------------------------------
// 3a) Weight repacks so the node-mix A fragments are n-contiguous (vector
//     loadable).  kconv (N,N,L,1) -> (L,N,N); comb (n,m) -> combT (m,n).
// ---------------------------------------------------------------------------
__global__ __launch_bounds__(256) void pack_kconv_kernel(
    const float* __restrict__ in, float* __restrict__ out) {
  const int idx = blockIdx.x * blockDim.x + threadIdx.x;
  if (idx >= kN * kN * kL) return;
  const int l = idx % kL;
  const int n = (idx / kL) % kN;
  const int m = idx / (kL * kN);
  out[((size_t)l * kN + m) * kN + n] = in[idx];
}

__global__ __launch_bounds__(256) void transpose_nn_kernel(
    const float* __restrict__ in, float* __restrict__ out) {
  const int idx = blockIdx.x * blockDim.x + threadIdx.x;
  if (idx >= kN * kN) return;
  const int n = idx % kN, m = idx / kN;
  out[(size_t)m * kN + n] = in[(size_t)n * kN + m];
}

// ---------------------------------------------------------------------------
// 3b) Node-mixing (dilated causal conv over the node axis, or dy_adj mix):
//     out[b,m,t,h] = sum_l sum_n W[l,m,n] * in[b,n,t - l*dil, h]
//     W is packed (taps, M, N) with n contiguous -> vector A-fragment loads.
//     16(node) x 16(feature) tile per wave, n stepped by 16 (4 WMMAs).
// ---------------------------------------------------------------------------
__global__ __launch_bounds__(256) void node_mix_kernel(
    const float* __restrict__ in, const float* __restrict__ W,
    int taps, int dil, float* __restrict__ out) {
  const int wave = (blockIdx.x * blockDim.x + threadIdx.x) >> 5;
  const int lane = threadIdx.x & 31;
  const int mt = kN >> 4;   // 16 node tiles
  const int ht = kH >> 4;   // 8 feature tiles
  const int total = kB * kT * mt * ht;
  if (wave >= total) return;

  int tmp = wave;
  const int hw = tmp % ht; tmp /= ht;
  const int mw = tmp % mt; tmp /= mt;
  const int t  = tmp % kT;
  const int b  = tmp / kT;
  const int m0 = mw << 4, h0 = hw << 4;
  const int r  = lane & 15;
  const int kh = (lane >> 4) << 1;
  const size_t nstr = (size_t)kT * kH;   // node stride in activation tensor

  v8f acc = {0.f, 0.f, 0.f, 0.f, 0.f, 0.f, 0.f, 0.f};
  for (int l = 0; l < taps; ++l) {
    const int ts = t - l * dil;
    if (ts < 0) continue;                      // causal zero padding (uniform)
    const float* Wrow = W + ((size_t)l * kN + (m0 + r)) * kN + kh;
    const float* Bl = in + (((size_t)b * kN) * kT + ts) * kH + h0 + r
                         + (size_t)kh * nstr;
    for (int k = 0; k < kN; k += 16) {
      const v2f a0 = *(const v2f*)(Wrow + k);
      const v2f a1 = *(const v2f*)(Wrow + k + 4);
      const v2f a2 = *(const v2f*)(Wrow + k + 8);
      const v2f a3 = *(const v2f*)(Wrow + k + 12);
      const float* bp = Bl + (size_t)k * nstr;
      v2f b0, b1, b2, b3;
      b0.x = bp[0];           b0.y = bp[nstr];
      b1.x = bp[4 * nstr];    b1.y = bp[5 * nstr];
      b2.x = bp[8 * nstr];    b2.y = bp[9 * nstr];
      b3.x = bp[12 * nstr];   b3.y = bp[13 * nstr];
      acc = wmma_f32(a0, b0, acc);
      acc = wmma_f32(a1, b1, acc);
      acc = wmma_f32(a2, b2, acc);
      acc = wmma_f32(a3, b3, acc);
    }
  }

  const int mbase = m0 + ((lane >> 4) << 3);
#pragma unroll
  for (int i = 0; i < 8; ++i) {
    out[(((size_t)b * kN + mbase + i) * kT + t) * kH + h0 + r] = acc[i];
  }
}

// ---------------------------------------------------------------------------
// 4) Attention per sequence: scores = QK^T/sqrt(H) -> softmax -> @V.
//    One 8-wave workgroup per sequence.  The V tile is staged into LDS with
//    CDNA5 async global->LDS copies (ASYNCcnt), overlapped with the QK^T
//    score WMMAs, and consumed from LDS by the attn@V WMMA pass.
// ---------------------------------------------------------------------------
__global__ __launch_bounds__(256) void attention_kernel(
    const float* __restrict__ q, const float* __restrict__ k,
    const float* __restrict__ v, float* __restrict__ ctb) {
  alignas(16) __shared__ float ss[kT * kT];   // 16 KB scores
  alignas(16) __shared__ float sv[kT * kH];   // 32 KB V tile
  const int seq = blockIdx.x;
  const int tid = threadIdx.x;
  const int wave = tid >> 5, lane = tid & 31;
  const int r = lane & 15, kh = (lane >> 4) << 1;
  const size_t base = (size_t)seq * kT * kH;
  const float rscale = 0.0883883476483184f;   // 1/sqrt(128)

  // Kick off async V -> LDS (8 x b128 per thread), overlapped with scores.
  {
    const unsigned svb = (unsigned)(uintptr_t)(void*)sv;
    const float* gv = v + base;
#pragma unroll
    for (int i = 0; i < 8; ++i) {
      const int idx = (i * 256 + tid) * 4;           // float index, 16B chunks
      const unsigned la = svb + (unsigned)(idx * 4); // LDS byte offset
      const float* ga = gv + idx;
      asm volatile("global_load_async_to_lds_b128 %0, %1, off"
                   :: "v"(la), "v"(ga) : "memory");
    }
  }

  // scores: 4x4 grid of 16x16 tiles, K=H=128
  for (int tile = wave; tile < 16; tile += 8) {
    const int t0 = (tile >> 2) << 4, s0 = (tile & 3) << 4;
    const float* qa = q + base + (size_t)(t0 + r) * kH + kh;
    const float* kb = k + base + (size_t)(s0 + r) * kH + kh;
    v8f acc = {0.f, 0.f, 0.f, 0.f, 0.f, 0.f, 0.f, 0.f};
    for (int kk = 0; kk < kH; kk += 16) {
      const v2f a0 = *(const v2f*)(qa + kk);
      const v2f a1 = *(const v2f*)(qa + kk + 4);
      const v2f a2 = *(const v2f*)(qa + kk + 8);
      const v2f a3 = *(const v2f*)(qa + kk + 12);
      const v2f b0 = *(const v2f*)(kb + kk);
      const v2f b1 = *(const v2f*)(kb + kk + 4);
      const v2f b2 = *(const v2f*)(kb + kk + 8);
      const v2f b3 = *(const v2f*)(kb + kk + 12);
      acc = wmma_f32(a0, b0, acc);
      acc = wmma_f32(a1, b1, acc);
      acc = wmma_f32(a2, b2, acc);
      acc = wmma_f32(a3, b3, acc);
    }
    const int tb = t0 + ((lane >> 4) << 3);
#pragma unroll
    for (int i = 0; i < 8; ++i) ss[(tb + i) * kT + s0 + r] = acc[i] * rscale;
  }
  __syncthreads();

  // row softmax (64 rows; thread i owns row i)
  if (tid < kT) {
    float mx = -1e30f;
    for (int s2 = 0; s2 < kT; ++s2) mx = fmaxf(mx, ss[tid * kT + s2]);
    float sum = 0.f;
    for (int s2 = 0; s2 < kT; ++s2) {
      const float e = __expf(ss[tid * kT + s2] - mx);
      ss[tid * kT + s2] = e;
      sum += e;
    }
    const float inv = 1.f / sum;
    for (int s2 = 0; s2 < kT; ++s2) ss[tid * kT + s2] *= inv;
  }
  // V tile must be resident in LDS before the second GEMM.
  asm volatile("s_wait_asynccnt 0x0" ::: "memory");
  __syncthreads();

  // ctb = attn(64x64) @ V(64x128): 4x8 grid of tiles, K=T=64 (V from LDS)
  for (int tile = wave; tile < 32; tile += 8) {
    const int t0 = (tile >> 3) << 4, h0 = (tile & 7) << 4;
    const float* aa = ss + (t0 + r) * kT + kh;
    const float* vb = sv + (size_t)kh * kH + h0 + r;
    v8f acc = {0.f, 0.f, 0.f, 0.f, 0.f, 0.f, 0.f, 0.f};
#pragma unroll 4
    for (int kk = 0; kk < kT; kk += 4) {
      const v2f a = *(const v2f*)(aa + kk);
      v2f b;
      b.x = vb[(size_t)kk * kH];
      b.y = vb[(size_t)(kk + 1) * kH];
      acc = wmma_f32(a, b, acc);
    }
    const int tb = t0 + ((lane >> 4) << 3);
#pragma unroll
    for (int i = 0; i < 8; ++i)
      ctb[base + (size_t)(tb + i) * kH + h0 + r] = acc[i];
  }
}

// ---------------------------------------------------------------------------
// 5) Residual add + LayerNorm (ddof=1, a*(x-m)/(s+1e-6)+b), one block per row.
// ---------------------------------------------------------------------------
__global__ __launch_bounds__(128) void add_ln_kernel(
    const float* __restrict__ a, const float* __restrict__ bsrc,
    const float* __restrict__ gain, const float* __restrict__ beta,
    float* __restrict__ out) {
  __shared__ float red[kH];
  const int row = blockIdx.x;
  const int h = threadIdx.x;
  const size_t idx = (size_t)row * kH + h;
  const float x = a[idx] + bsrc[idx];

  red[h] = x; __syncthreads();
  for (int off = kH >> 1; off > 0; off >>= 1) {
    if (h < off) red[h] += red[h + off];
    __syncthreads();
  }
  const float m = red[0] * (1.f / kH);
  __syncthreads();

  const float d = x - m;
  red[h] = d * d; __syncthreads();
  for (int off = kH >> 1; off > 0; off >>= 1) {
    if (h < off) red[h] += red[h + off];
    __syncthreads();
  }
  const float s = sqrtf(red[0] / (kH - 1));
  out[idx] = gain[h] * d / (s + 1e-6f) + beta[h];
}

// ---------------------------------------------------------------------------
// 6) Final: last timestep @ fc_out + softmax over nodes per batch.
// ---------------------------------------------------------------------------
__global__ __launch_bounds__(256) void final_kernel(
    const float* __restrict__ xf, const float* __restrict__ w,
    const float* __restrict__ bptr, float* __restrict__ out) {
  __shared__ float tmp[kN];
  const int b = blockIdx.x, n = threadIdx.x;
  const float* xp = xf + (((size_t)(b * kN + n)) * kT + (kT - 1)) * kH;
  float acc = bptr[0];
#pragma unroll 8
  for (int h = 0; h < kH; ++h) acc += xp[h] * w[h];

  tmp[n] = acc; __syncthreads();
  for (int off = kN >> 1; off > 0; off >>= 1) {
    if (n < off) tmp[n] = fmaxf(tmp[n], tmp[n + off]);
    __syncthreads();
  }
  const float mx = tmp[0]; __syncthreads();
  const float e = __expf(acc - mx);
  tmp[n] = e; __syncthreads();
  for (int off = kN >> 1; off > 0; off >>= 1) {
    if (n < off) tmp[n] += tmp[n + off];
    __syncthreads();
  }
  out[(size_t)b * kN + n] = e / tmp[0];
}

// ---------------------------------------------------------------------------
// Host-side orchestration
// ---------------------------------------------------------------------------
static void launch_gemm(hipStream_t st, const float* A, int lda,
                        const float* Bm, int sbj, int sbk, const float* bias,
                        float* C, int ldc, int M, int Nc, int K) {
  const int tiles = (M / 16) * (Nc / 16);
  const int blocks = (tiles + 7) / 8;
  if (sbk == 1) {
    wmma_gemm_kernel<true><<<blocks, 256, 0, st>>>(A, lda, Bm, sbj, sbk, bias,
                                                   C, ldc, M, Nc, K);
  } else {
    wmma_gemm_kernel<false><<<blocks, 256, 0, st>>>(A, lda, Bm, sbj, sbk, bias,
                                                    C, ldc, M, Nc, K);
  }
}

static void launch_node_mix(hipStream_t st, const float* in, const float* Wp,
                            int taps, int dil, float* out) {
  const int tiles = kB * kT * (kN / 16) * (kH / 16);   // 32768
  node_mix_kernel<<<(tiles + 7) / 8, 256, 0, st>>>(in, Wp, taps, dil, out);
}

extern "C" void kernel_launch(void* const* d_in, const int* in_sizes, int n_in,
                              void* d_out, int out_size, void* d_ws, size_t ws_size,
                              hipStream_t stream) {
  (void)in_sizes; (void)n_in; (void)out_size; (void)ws_size;
  // Input order: x, ind_adj (unused), sec_adj, then 22 tensors per FSTB layer
  // (in_w,in_b,fwr,fwi,fbr,fbi,qw,qb,kw,kb,vw,vb,kconv,vconv,ln1_a,ln1_b,
  //  gw,dy_adj,ow,ob,ln2_a,ln2_b) x3, then fc_out_w, fc_out_b.
  const float* x       = (const float*)d_in[0];
  const float* sec_adj = (const float*)d_in[2];

  float* ws = (float*)d_ws;
  const size_t SLOT = (size_t)kBN * kT * kH;   // 8.39M floats
  float* S0 = ws + 0 * SLOT;  // spectral output
  float* S1 = ws + 1 * SLOT;  // xf / y
  float* S2 = ws + 2 * SLOT;  // q / ln1 out
  float* S3 = ws + 3 * SLOT;  // k / ctb
  float* S4 = ws + 4 * SLOT;  // v / y2
  float* S5 = ws + 5 * SLOT;  // kc / ow-out
  float* S6 = ws + 6 * SLOT;  // vc / layer output
  // small scratch region inside slot 7
  float* comb  = ws + 7 * SLOT;                       // 256x256
  float* combT = comb  + (size_t)kN * kN;             // 256x256 transposed
  float* kpack = combT + (size_t)kN * kN;             // (L,N,N) packed kconv
  float* vpack = kpack + (size_t)kL * kN * kN;        // (L,N,N) packed vconv

  const int dils[3] = {1, 2, 4};
  const float* cur = x;
  int din = kP;

  for (int layer = 0; layer < 3; ++layer) {
    const int pb = 3 + layer * 22;
    auto prm = [&](int i) { return (const float*)d_in[pb + i]; };

    // repack conv weights into n-contiguous (L,M,N) for vector A-fragments
    const int pk = kN * kN * kL;
    pack_kconv_kernel<<<(pk + 255) / 256, 256, 0, stream>>>(prm(12), kpack);
    pack_kconv_kernel<<<(pk + 255) / 256, 256, 0, stream>>>(prm(13), vpack);

    // spectral low-pass
    spectral_kernel<<<kBN, 128, 0, stream>>>(cur, S0, prm(2), prm(3), prm(4), prm(5), din);
    // xf = xlow @ in_w^T + in_b
    launch_gemm(stream, S0, din, prm(0), din, 1, prm(1), S1, kH, kBN * kT, kH, din);
    // q, k, v projections
    launch_gemm(stream, S1, kH, prm(6),  kH, 1, prm(7),  S2, kH, kBN * kT, kH, kH);
    launch_gemm(stream, S1, kH, prm(8),  kH, 1, prm(9),  S3, kH, kBN * kT, kH, kH);
    launch_gemm(stream, S1, kH, prm(10), kH, 1, prm(11), S4, kH, kBN * kT, kH, kH);
    // causal dilated node-mixing convs (packed weights)
    launch_node_mix(stream, S3, kpack, kL, dils[layer], S5);
    launch_node_mix(stream, S4, vpack, kL, dils[layer], S6);
    // attention: ctb -> S3
    attention_kernel<<<kBN, 256, 0, stream>>>(S2, S5, S6, S3);
    // xf2 = LN(xf + ctb) -> S2
    add_ln_kernel<<<kBN * kT, 128, 0, stream>>>(S1, S3, prm(14), prm(15), S2);
    // y = xf2 @ gw  (out[.,j] = sum_h xf2[.,h]*gw[h,j] -> sbj=1, sbk=H)
    launch_gemm(stream, S2, kH, prm(16), 1, kH, nullptr, S1, kH, kBN * kT, kH, kH);
    // comb = sec_adj @ dy_adj  (plain A@B -> sbj=1, sbk=N), then transpose
    launch_gemm(stream, sec_adj, kN, prm(17), 1, kN, nullptr, comb, kN, kN, kN, kN);
    transpose_nn_kernel<<<(kN * kN + 255) / 256, 256, 0, stream>>>(comb, combT);
    // y2[b,m,t,h] = sum_n combT[m,n] * y[b,n,t,h]
    launch_node_mix(stream, S1, combT, 1, 1, S4);
    // ow projection and second residual LN
    launch_gemm(stream, S4, kH, prm(18), kH, 1, prm(19), S5, kH, kBN * kT, kH, kH);
    add_ln_kernel<<<kBN * kT, 128, 0, stream>>>(S4, S5, prm(20), prm(21), S6);

    cur = S6;
    din = kH;
  }

  const float* fcw = (const float*)d_in[3 + 3 * 22];
  const float* fcb = (const float*)d_in[3 + 3 * 22 + 1];
  final_kernel<<<kB, 256, 0, stream>>>(S6, fcw, fcb, (float*)d_out);
}